// SpikesToTimesDecoder_54606214201659
// MI455X (gfx1250) — compile-verified
//
#include <hip/hip_runtime.h>

// SpikesToTimesDecoder: first SPIKE_COUNT(=3) spike times along T per (b,n)
// channel, ascending, inf-padded, scaled by dt=0.001.
// Input  [T=512, B=64, N=2048] float32 (0/1 raster)
// Output [3, B, N] float32
//
// Memory-bound scan (268 MB in, 1.5 MB out; ~11.5us full-scan roofline at
// 23.3 TB/s). gfx1250 path: async global->LDS staging pipeline
// (global_load_async_to_lds_b128 th:TH_LOAD_NT / s_wait_asynccnt) with an
// 8-deep chunk pipeline (14 KB in flight per wave, ~14 MB machine-wide across
// the 1024 structurally-available waves -> enough to cover HBM latency at
// 23.3 TB/s), plus wave32-uniform early exit once all 128 channels owned by a
// wave have found 3 spikes (~3x traffic reduction at 5% spike rate).

namespace {
constexpr int T_DIM   = 512;
constexpr int B_DIM   = 64;
constexpr int N_DIM   = 2048;
constexpr int BN      = B_DIM * N_DIM;     // 131072 channels
constexpr int KSP     = 3;
constexpr float DT_F  = 0.001f;

constexpr int THREADS = 128;               // 4 waves/block
constexpr int CPT     = 4;                 // channels per thread (16B quad)
constexpr int CPB     = THREADS * CPT;     // 512 channels per block
constexpr int CH      = 4;                 // timesteps per chunk
constexpr int NCHUNK  = T_DIM / CH;        // 128
constexpr int DEPTH   = 8;                 // chunks in flight (power of 2)
constexpr int SLICE   = CPB;               // floats per LDS slice (512)
} // namespace

typedef __attribute__((address_space(3))) float lds_float;

__global__ __launch_bounds__(THREADS) void spike_first3_kernel(
    const float* __restrict__ in, float* __restrict__ out) {
  // DEPTH * CH * 512 floats = 64 KB LDS (<= 5 blocks/WGP of the 320 KB pool)
  __shared__ __align__(16) float smem[DEPTH * CH * SLICE];

  const int tid  = threadIdx.x;
  const int chan = blockIdx.x * CPB + tid * CPT;   // first of 4 channels
  const float* g0 = in + chan;
  lds_float* lbase = (lds_float*)smem + tid * CPT;

  // Per-channel state: spike count + up to 3 spike time indices.
  int c0 = 0, c1 = 0, c2 = 0, c3 = 0;
  int q00 = 0, q01 = 0, q02 = 0;
  int q10 = 0, q11 = 0, q12 = 0;
  int q20 = 0, q21 = 0, q22 = 0;
  int q30 = 0, q31 = 0, q32 = 0;

  // Issue one chunk: CH async b128 loads (one per timestep slice).
  // Each lane moves its own 16B strip global->LDS; ASYNCcnt += CH per chunk.
  // NT hint: 268 MB streamed once > 192 MB L2, don't thrash it.
  auto issue_chunk = [&](int c) {
    const float* gp = g0 + (long)c * CH * BN;
    lds_float* lp = lbase + (c & (DEPTH - 1)) * (CH * SLICE);
#pragma unroll
    for (int s = 0; s < CH; ++s) {
      asm volatile("global_load_async_to_lds_b128 %0, %1, off th:TH_LOAD_NT"
                   :
                   : "v"(lp + s * SLICE), "v"(gp + (long)s * BN)
                   : "memory");
    }
  };

#pragma unroll
  for (int p = 0; p < DEPTH; ++p) issue_chunk(p);

  for (int c = 0; c < NCHUNK; ++c) {
    // Wait until chunk c has fully landed in LDS. Steady state keeps
    // (DEPTH-1)*CH = 28 newer async loads outstanding; drain fully in the
    // (rare, early-exit-dominated) tail.
    if (c + DEPTH <= NCHUNK) {
      asm volatile("s_wait_asynccnt 28" ::: "memory");
    } else {
      asm volatile("s_wait_asynccnt 0" ::: "memory");
    }

    const float* sp = smem + (c & (DEPTH - 1)) * (CH * SLICE) + tid * CPT;
#pragma unroll
    for (int s = 0; s < CH; ++s) {
      const int t = c * CH + s;
      const float4 v = *(const float4*)(sp + s * SLICE);
      { const bool f = v.x > 0.0f;
        q00 = (f && c0 == 0) ? t : q00;
        q01 = (f && c0 == 1) ? t : q01;
        q02 = (f && c0 == 2) ? t : q02;
        c0 += (f && c0 < KSP) ? 1 : 0; }
      { const bool f = v.y > 0.0f;
        q10 = (f && c1 == 0) ? t : q10;
        q11 = (f && c1 == 1) ? t : q11;
        q12 = (f && c1 == 2) ? t : q12;
        c1 += (f && c1 < KSP) ? 1 : 0; }
      { const bool f = v.z > 0.0f;
        q20 = (f && c2 == 0) ? t : q20;
        q21 = (f && c2 == 1) ? t : q21;
        q22 = (f && c2 == 2) ? t : q22;
        c2 += (f && c2 < KSP) ? 1 : 0; }
      { const bool f = v.w > 0.0f;
        q30 = (f && c3 == 0) ? t : q30;
        q31 = (f && c3 == 1) ? t : q31;
        q32 = (f && c3 == 2) ? t : q32;
        c3 += (f && c3 < KSP) ? 1 : 0; }
    }

    // Wave-uniform early exit: all 128 channels owned by this wave done.
    const bool done = (c0 >= KSP) & (c1 >= KSP) & (c2 >= KSP) & (c3 >= KSP);
    if (__all(done)) break;

    if (c + DEPTH < NCHUNK) {
      // Buffer (c & 7) was just read; make sure the ds reads retired before
      // the async engine is allowed to overwrite it.
      asm volatile("s_wait_dscnt 0" ::: "memory");
      issue_chunk(c + DEPTH);
    }
  }

  // Emit times (idx * dt), inf-padded. Coalesced b128 stores per k-plane.
  const float INF = __builtin_inff();
  float4 o;
  o.x = (c0 > 0) ? (float)q00 * DT_F : INF;
  o.y = (c1 > 0) ? (float)q10 * DT_F : INF;
  o.z = (c2 > 0) ? (float)q20 * DT_F : INF;
  o.w = (c3 > 0) ? (float)q30 * DT_F : INF;
  *(float4*)(out + 0 * BN + chan) = o;

  o.x = (c0 > 1) ? (float)q01 * DT_F : INF;
  o.y = (c1 > 1) ? (float)q11 * DT_F : INF;
  o.z = (c2 > 1) ? (float)q21 * DT_F : INF;
  o.w = (c3 > 1) ? (float)q31 * DT_F : INF;
  *(float4*)(out + 1 * BN + chan) = o;

  o.x = (c0 > 2) ? (float)q02 * DT_F : INF;
  o.y = (c1 > 2) ? (float)q12 * DT_F : INF;
  o.z = (c2 > 2) ? (float)q22 * DT_F : INF;
  o.w = (c3 > 2) ? (float)q32 * DT_F : INF;
  *(float4*)(out + 2 * BN + chan) = o;
}

extern "C" void kernel_launch(void* const* d_in, const int* in_sizes, int n_in,
                              void* d_out, int out_size, void* d_ws, size_t ws_size,
                              hipStream_t stream) {
  (void)in_sizes; (void)n_in; (void)out_size; (void)d_ws; (void)ws_size;
  const float* in = (const float*)d_in[0];
  float* out = (float*)d_out;
  dim3 grid(BN / CPB);   // 256 blocks
  dim3 block(THREADS);   // 128 threads = 4 waves (wave32)
  hipLaunchKernelGGL(spike_first3_kernel, grid, block, 0, stream, in, out);
}